// GCONV_89953795047573
// MI455X (gfx1250) — compile-verified
//
#include <hip/hip_runtime.h>

// CDNA5 / gfx1250, wave32.
typedef __attribute__((ext_vector_type(2))) float v2f;
typedef __attribute__((ext_vector_type(8))) float v8f;

#define TPB 256

// ---------------- utility kernels ----------------

__global__ void zero_kernel(float* __restrict__ p, int n) {
  int i = blockIdx.x * blockDim.x + threadIdx.x;
  if (i < n) p[i] = 0.0f;
}

// Unweighted degree accumulation (counts as floats so the norm transform is in-place).
__global__ void deg_kernel(const int* __restrict__ src, const int* __restrict__ dst,
                           float* __restrict__ deg_out, float* __restrict__ deg_in, int E) {
  int e = blockIdx.x * blockDim.x + threadIdx.x;
  if (e >= E) return;
  unsafeAtomicAdd(&deg_out[src[e]], 1.0f);   // hardware global_atomic_add_f32, L2-resident
  unsafeAtomicAdd(&deg_in[dst[e]], 1.0f);
}

// deg -> clamp(deg,1)^-1/2, in place for both arrays.
__global__ void norm_kernel(float* __restrict__ a, float* __restrict__ b, int n) {
  int i = blockIdx.x * blockDim.x + threadIdx.x;
  if (i >= n) return;
  a[i] = rsqrtf(fmaxf(a[i], 1.0f));
  b[i] = rsqrtf(fmaxf(b[i], 1.0f));
}

// Collapse edge MLP:  w_eff[16] = W_l1 @ W_l2 ,  slot 16 = b_l1@W_l2 + b_l2.
// (No nonlinearity between lin1/lin2 in the reference, so this is exact.)
__global__ void weff_kernel(const float* __restrict__ W_l1, const float* __restrict__ b_l1,
                            const float* __restrict__ W_l2, const float* __restrict__ b_l2,
                            float* __restrict__ weff) {
  int i = threadIdx.x;
  if (i < 16) {
    float s = 0.0f;
#pragma unroll
    for (int j = 0; j < 8; ++j) s += W_l1[i * 8 + j] * W_l2[j];
    weff[i] = s;
  } else if (i == 16) {
    float s = b_l2[0];
#pragma unroll
    for (int j = 0; j < 8; ++j) s += b_l1[j] * W_l2[j];
    weff[16] = s;
  }
}

// ew[e] = edges[e] . w_eff + b_eff   (16-wide dot, streams 64B/edge from HBM)
__global__ void ew_kernel(const float* __restrict__ edges, const float* __restrict__ weff,
                          float* __restrict__ ew, int E) {
  int e = blockIdx.x * blockDim.x + threadIdx.x;
  if (e >= E) return;
  const float4* ep = (const float4*)(edges + (size_t)e * 16);
  float acc = weff[16];
#pragma unroll
  for (int i = 0; i < 4; ++i) {
    float4 v = ep[i];
    acc += v.x * weff[i * 4 + 0] + v.y * weff[i * 4 + 1] +
           v.z * weff[i * 4 + 2] + v.w * weff[i * 4 + 3];
  }
  ew[e] = acc;
}

// ---------------- dense GEMM via V_WMMA_F32_16X16X4_F32 ----------------
// x[r,:] = a_row(r) @ W  where W is 32x32 row-major.
//   MODE 0 (layer 1): a_row(r)[k] = feat[r,k] * out_norm[r]
//   MODE 1 (layer 2): a_row(r)[k] = relu(feat[r,k]*in_norm[r] + b_prev[k]) * out_norm[r]
// MODE is a template parameter -> branch-free fully-unrolled inner loop.
// One wave per 16-row tile; two 16x16 C accumulators cover all 32 output columns.
// A-fragment (ISA 7.12.2, 32-bit 16x4): lane L -> A[L%16][kc + 2*(L/16) + {0,1}]
// B-fragment (symmetric):               lane L -> B[kc + 2*(L/16) + {0,1}][n0 + L%16]
// C/D:                                  lane L, vgpr i -> C[i + 8*(L/16)][n0 + L%16]
template <int MODE>
__global__ void gemm_wmma_kernel(const float* __restrict__ feat,
                                 const float* __restrict__ W,
                                 const float* __restrict__ out_norm,
                                 const float* __restrict__ in_norm,
                                 const float* __restrict__ b_prev,
                                 float* __restrict__ xout,
                                 int Nn) {
  const int wave = threadIdx.x >> 5;
  const int lane = threadIdx.x & 31;
  const int tile = blockIdx.x * (blockDim.x >> 5) + wave;
  const int r0 = tile * 16;
  if (r0 >= Nn) return;                       // wave-uniform exit: EXEC stays all-1s for WMMA
  const int half = lane >> 4;
  const int l16 = lane & 15;
  int row = r0 + l16;
  if (row >= Nn) row = Nn - 1;                // clamp loads (stores guarded below)

  const float on = out_norm[row];
  const float inn = (MODE == 1) ? in_norm[row] : 0.0f;

  v8f acc0 = {};
  v8f acc1 = {};
#pragma unroll
  for (int kc = 0; kc < 8; ++kc) {
    const int k = kc * 4 + half * 2;          // this lane's two K slots
    float a0 = feat[row * 32 + k];
    float a1 = feat[row * 32 + k + 1];
    if (MODE == 1) {
      a0 = fmaxf(a0 * inn + b_prev[k], 0.0f);
      a1 = fmaxf(a1 * inn + b_prev[k + 1], 0.0f);
    }
    v2f a;  a.x = a0 * on;               a.y = a1 * on;
    v2f b0; b0.x = W[k * 32 + l16];      b0.y = W[(k + 1) * 32 + l16];
    v2f b1; b1.x = W[k * 32 + 16 + l16]; b1.y = W[(k + 1) * 32 + 16 + l16];
    acc0 = __builtin_amdgcn_wmma_f32_16x16x4_f32(false, a, false, b0, (short)0, acc0, false, false);
    acc1 = __builtin_amdgcn_wmma_f32_16x16x4_f32(false, a, false, b1, (short)0, acc1, false, false);
  }

#pragma unroll
  for (int i = 0; i < 8; ++i) {
    const int rr = r0 + half * 8 + i;
    if (rr < Nn) {
      xout[rr * 32 + l16] = acc0[i];
      xout[rr * 32 + 16 + l16] = acc1[i];
    }
  }
}

// ---------------- edge scatter: agg[dst] += x[src] * ew ----------------
// 8 threads per edge, float4 gather from x (L2-resident), 4 hw f32 atomics each.
__global__ void scatter_kernel(const float* __restrict__ x, const float* __restrict__ ew,
                               const int* __restrict__ src, const int* __restrict__ dst,
                               float* __restrict__ agg, int E) {
  const int t = blockIdx.x * blockDim.x + threadIdx.x;
  const int e = t >> 3;
  const int c = t & 7;
  if (e >= E) return;
  const float w = ew[e];
  const int s = src[e];
  const int d = dst[e];
  const float4 v = ((const float4*)(x + (size_t)s * 32))[c];
  float* ad = agg + (size_t)d * 32 + c * 4;
  unsafeAtomicAdd(ad + 0, v.x * w);
  unsafeAtomicAdd(ad + 1, v.y * w);
  unsafeAtomicAdd(ad + 2, v.z * w);
  unsafeAtomicAdd(ad + 3, v.w * w);
}

// ---------------- final epilogue: out = relu(agg * in_norm + b) ----------------
__global__ void final_kernel(const float* __restrict__ agg, const float* __restrict__ in_norm,
                             const float* __restrict__ b, float* __restrict__ out, int Nn) {
  const int t = blockIdx.x * blockDim.x + threadIdx.x;  // one float4 per thread
  if (t >= Nn * 8) return;
  const int node = t >> 3;
  const int c4 = (t & 7) * 4;
  const float inn = in_norm[node];
  float4 v = ((const float4*)agg)[t];
  float4 r;
  r.x = fmaxf(v.x * inn + b[c4 + 0], 0.0f);
  r.y = fmaxf(v.y * inn + b[c4 + 1], 0.0f);
  r.z = fmaxf(v.z * inn + b[c4 + 2], 0.0f);
  r.w = fmaxf(v.w * inn + b[c4 + 3], 0.0f);
  ((float4*)out)[t] = r;
}

// ---------------- host orchestration ----------------

extern "C" void kernel_launch(void* const* d_in, const int* in_sizes, int n_in,
                              void* d_out, int out_size, void* d_ws, size_t ws_size,
                              hipStream_t stream) {
  const float* inputs = (const float*)d_in[0];   // [N,32]
  const float* edges  = (const float*)d_in[1];   // [E,16]
  const int*   src    = (const int*)d_in[2];     // [E]
  const int*   dst    = (const int*)d_in[3];     // [E]
  const float* W_l1   = (const float*)d_in[4];   // [16,8]
  const float* b_l1   = (const float*)d_in[5];   // [8]
  const float* W_l2   = (const float*)d_in[6];   // [8,1]
  const float* b_l2   = (const float*)d_in[7];   // [1]
  const float* W_c1   = (const float*)d_in[8];   // [32,32]
  const float* b_c1   = (const float*)d_in[9];   // [32]
  const float* W_c2   = (const float*)d_in[10];  // [32,32]
  const float* b_c2   = (const float*)d_in[11];  // [32]
  float* out = (float*)d_out;

  const int N = in_sizes[0] / 32;
  const int E = in_sizes[2];

  // Workspace layout (floats), all 16B-aligned for float4 access.
  float* ws = (float*)d_ws;
  float* out_norm = ws;                 // [N]   (deg accumulator, then norm)
  float* in_norm  = ws + N;             // [N]
  float* weff     = ws + 2 * N;         // [32]  (16 weights + 1 bias)
  float* ew       = weff + 32;          // [E]
  float* xbuf     = ew + E;             // [N*32]  x1 then x2
  float* aggbuf   = xbuf + (size_t)N * 32;  // [N*32]  agg1 then agg2

  const int gE   = (E + TPB - 1) / TPB;
  const int gN   = (N + TPB - 1) / TPB;
  const int gNf  = (N * 32 + TPB - 1) / TPB;
  const int gSc  = (E * 8 + TPB - 1) / TPB;
  const int gFin = (N * 8 + TPB - 1) / TPB;
  const int tiles = (N + 15) / 16;
  const int gGemm = (tiles + (TPB / 32) - 1) / (TPB / 32);

  // Degrees -> norms
  zero_kernel<<<(2 * N + TPB - 1) / TPB, TPB, 0, stream>>>(out_norm, 2 * N);
  deg_kernel<<<gE, TPB, 0, stream>>>(src, dst, out_norm, in_norm, E);
  norm_kernel<<<gN, TPB, 0, stream>>>(out_norm, in_norm, N);

  // Collapsed edge MLP
  weff_kernel<<<1, 32, 0, stream>>>(W_l1, b_l1, W_l2, b_l2, weff);
  ew_kernel<<<gE, TPB, 0, stream>>>(edges, weff, ew, E);

  // Layer 1: x1 = (inputs * out_norm) @ W_c1 ; agg1 = scatter(x1)
  gemm_wmma_kernel<0><<<gGemm, TPB, 0, stream>>>(inputs, W_c1, out_norm, in_norm, b_c1,
                                                 xbuf, N);
  zero_kernel<<<gNf, TPB, 0, stream>>>(aggbuf, N * 32);
  scatter_kernel<<<gSc, TPB, 0, stream>>>(xbuf, ew, src, dst, aggbuf, E);

  // Layer 2: x2 = (relu(agg1*in_norm + b_c1) * out_norm) @ W_c2 (epilogue fused into A load)
  gemm_wmma_kernel<1><<<gGemm, TPB, 0, stream>>>(aggbuf, W_c2, out_norm, in_norm, b_c1,
                                                 xbuf, N);
  zero_kernel<<<gNf, TPB, 0, stream>>>(aggbuf, N * 32);
  scatter_kernel<<<gSc, TPB, 0, stream>>>(xbuf, ew, src, dst, aggbuf, E);

  // out = relu(agg2 * in_norm + b_c2)
  final_kernel<<<gFin, TPB, 0, stream>>>(aggbuf, in_norm, b_c2, out, N);
  (void)n_in; (void)out_size; (void)ws_size;
}